// QueryAndGroup_62835371540837
// MI455X (gfx1250) — compile-verified
//
#include <hip/hip_runtime.h>

typedef __attribute__((ext_vector_type(2))) float v2f;
typedef __attribute__((ext_vector_type(8))) float v8f;

#define BATCH     4
#define NPTS      16384
#define MQ        2048
#define KSAMP     32
#define CFEAT     64
#define COUT      67          // 3 + 64
#define RADIUS2   0.16f       // 0.4^2

// ---------------------------------------------------------------------------
// Kernel 1: ball-query bitmask via WMMA distance tiles.
//
// dist^2 = |p|^2 + |q|^2 - 2 p.q with ONE V_WMMA_F32_16X16X4_F32 per tile:
//   A (16x4, rows = points)  : (-2px, -2py, -2pz, |p|^2)
//   B (4x16, cols = queries) : ( qx,   qy,   qz,  1    )
//   C preloaded with |q|^2 (uniform down each column)
//
// D layout (wave32): lane L holds query column L%16 and point rows
// 8*(L/16)+i in VGPR i -> each lane packs 8 predicate bits for 8
// consecutive point indices into one mask byte (no cross-lane traffic).
// Bytes staged in LDS, flushed with GLOBAL_STORE_ASYNC_FROM_LDS_B128.
//
// grid = BATCH * (MQ/16) blocks, 128 threads (4 waves); wave w owns the
// point range [w*4096, (w+1)*4096).
// ---------------------------------------------------------------------------
__global__ __launch_bounds__(128)
void ball_query_mask_kernel(const float* __restrict__ points,   // (B,N,3)
                            const float* __restrict__ queries,  // (B,M,3)
                            unsigned int* __restrict__ mask)    // (B*M, N/32)
{
    const int lane  = threadIdx.x & 31;
    const int wave  = threadIdx.x >> 5;
    const int batch = blockIdx.x >> 7;          // / (MQ/16)
    const int m0    = (blockIdx.x & 127) << 4;  // query tile base
    const int hf    = lane >> 4;                // 0: lanes 0-15, 1: lanes 16-31
    const int l16   = lane & 15;

    // --- B matrix (queries, fixed for whole kernel) + C = |q|^2 ------------
    const float* qp = queries + ((size_t)batch * MQ + (m0 + l16)) * 3;
    const float qx = qp[0], qy = qp[1], qz = qp[2];
    v2f bmat;
    bmat.x = hf ? qz   : qx;    // K=2 : K=0
    bmat.y = hf ? 1.0f : qy;    // K=3 : K=1
    const float qn = qx * qx + qy * qy + qz * qz;
    v8f cacc;
#pragma unroll
    for (int i = 0; i < 8; ++i) cacc[i] = qn;

    // per-wave LDS staging: 16 query rows x 512 mask bytes (n-range 4096)
    __shared__ alignas(16) unsigned char smask[4][16][512];

    const float* pbase   = points + (size_t)batch * NPTS * 3;
    const int    n_begin = wave * (NPTS / 4);
    const int    n_end   = n_begin + (NPTS / 4);

#pragma unroll 4                        // 4 WMMAs (64 points) per iteration
    for (int n0 = n_begin; n0 < n_end; n0 += 16) {
        const float* pp = pbase + (size_t)(n0 + l16) * 3;
        const float px = pp[0], py = pp[1], pz = pp[2];
        const float pn = px * px + py * py + pz * pz;

        // branch-free operand select (keeps EXEC all-ones around WMMA)
        v2f amat;
        amat.x = hf ? (-2.0f * pz) : (-2.0f * px);   // K=2 : K=0
        amat.y = hf ? pn           : (-2.0f * py);   // K=3 : K=1

        v8f d = __builtin_amdgcn_wmma_f32_16x16x4_f32(
            /*neg_a=*/false, amat, /*neg_b=*/false, bmat,
            /*c_mod=*/(short)0, cacc, /*reuse_a=*/false, /*reuse_b=*/false);

        unsigned int byte = 0u;
#pragma unroll
        for (int i = 0; i < 8; ++i)
            byte |= (d[i] <= RADIUS2 ? 1u : 0u) << i;

        // lane covers points n0 + 8*hf + {0..7} for query column m0+l16
        smask[wave][l16][((n0 - n_begin) >> 3) + hf] = (unsigned char)byte;
    }

    __syncthreads();   // DS complete before async engine reads LDS

    // --- flush: LDS -> global without VGPR round-trip (ASYNCcnt path) -----
    // 16 rows x 512 B per wave = 512 x 16B chunks; 16 b128 stores per lane.
    const unsigned lds0  = (unsigned)(uintptr_t)&smask[wave][0][0];
    const unsigned long long mbase = (unsigned long long)(uintptr_t)mask;
    const unsigned rowg0 = (unsigned)(((unsigned)batch * MQ + m0) * (NPTS / 8));
    const unsigned woff  = (unsigned)(n_begin >> 3);   // wave's byte offset in row

    for (int i = lane; i < 512; i += 32) {
        const int r  = i >> 5;                 // query row within tile
        const int ch = (i & 31) << 4;          // 16B chunk within row
        const unsigned goff = rowg0 + (unsigned)r * (NPTS / 8) + woff + ch;
        const unsigned loff = lds0 + (unsigned)(r * 512) + ch;
        asm volatile("global_store_async_from_lds_b128 %0, %1, %2"
                     :: "v"(goff), "v"(loff), "s"(mbase) : "memory");
    }
    asm volatile("s_wait_asynccnt 0" ::: "memory");
}

// ---------------------------------------------------------------------------
// Kernel 2: ordered first-32 selection from the bitmask + gather/group.
// One wave32 per query point (4 waves / block).
// ---------------------------------------------------------------------------
__global__ __launch_bounds__(128)
void select_group_kernel(const float* __restrict__ points,    // (B,N,3)
                         const float* __restrict__ queries,   // (B,M,3)
                         const float* __restrict__ features,  // (B,C,N)
                         const unsigned int* __restrict__ mask,
                         float* __restrict__ out)             // (B,67,M,K)
{
    const int lane = threadIdx.x & 31;
    const int wave = threadIdx.x >> 5;
    const int bm   = blockIdx.x * 4 + wave;     // flat (batch, m)
    const int batch = bm >> 11;                 // / MQ
    const int m     = bm & (MQ - 1);

    __shared__ int idxbuf[4][KSAMP];

    const unsigned int* row = mask + (size_t)bm * (NPTS / 32);

    int base = 0;  // wave-uniform running count of valid neighbors
    for (int c = 0; c < NPTS / 32 && base < KSAMP; c += 32) {
        __builtin_prefetch(&row[c + 32 + lane], 0, 1);  // global_prefetch_b8
        unsigned int w = row[c + lane];
        int cnt = __popc(w);

        // inclusive prefix sum across the wave (lane permutes)
        int pre = cnt;
#pragma unroll
        for (int d = 1; d < 32; d <<= 1) {
            int t = __shfl_up(pre, d, 32);
            if (lane >= d) pre += t;
        }
        const int total = __shfl(pre, 31, 32);
        int slot = base + (pre - cnt);          // exclusive prefix + base

        while (w != 0u && slot < KSAMP) {       // append set bits in order
            const int j = __ffs(w) - 1;
            w &= w - 1u;
            idxbuf[wave][slot] = ((c + lane) << 5) + j;
            ++slot;
        }
        base += total;
    }
    __syncthreads();

    // padding semantics: pad with last valid index, or 0 if none found.
    const int cnt = base;                       // exact when < KSAMP
    int myidx;
    if (cnt == 0) {
        myidx = 0;
    } else {
        const int eff   = cnt < KSAMP ? cnt : KSAMP;
        const int lastv = idxbuf[wave][eff - 1];
        myidx = (lane < cnt) ? idxbuf[wave][lane] : lastv;
    }

    // --- gather + recenter + concat; lane == K dim => coalesced stores -----
    const float* pp = points  + ((size_t)batch * NPTS + myidx) * 3;
    const float* qp = queries + ((size_t)batch * MQ   + m) * 3;   // uniform
    const size_t ch = (size_t)MQ * KSAMP;                          // 65536
    float* o = out + (((size_t)batch * COUT) * MQ + m) * KSAMP + lane;

    o[0]      = pp[0] - qp[0];
    o[ch]     = pp[1] - qp[1];
    o[2 * ch] = pp[2] - qp[2];

    const float* f = features + ((size_t)batch * CFEAT) * NPTS + myidx;
#pragma unroll 4
    for (int c = 0; c < CFEAT; ++c)
        o[(size_t)(3 + c) * ch] = f[(size_t)c * NPTS];
}

// ---------------------------------------------------------------------------
extern "C" void kernel_launch(void* const* d_in, const int* in_sizes, int n_in,
                              void* d_out, int out_size, void* d_ws, size_t ws_size,
                              hipStream_t stream)
{
    const float* points   = (const float*)d_in[0];   // (4,16384,3)
    const float* queries  = (const float*)d_in[1];   // (4,2048,3)
    const float* features = (const float*)d_in[2];   // (4,64,16384)
    float*       out      = (float*)d_out;           // (4,67,2048,32)

    // bitmask workspace: B*M rows x N bits = 16 MiB (lives in the 192MB L2)
    unsigned int* mask = (unsigned int*)d_ws;

    ball_query_mask_kernel<<<dim3(BATCH * (MQ / 16)), dim3(128), 0, stream>>>(
        points, queries, mask);

    select_group_kernel<<<dim3((BATCH * MQ) / 4), dim3(128), 0, stream>>>(
        points, queries, features, mask, out);
}